// LiteMLA_27728308863814
// MI455X (gfx1250) — compile-verified
//
#include <hip/hip_runtime.h>
#include <math.h>

// ---------------- problem constants ----------------
#define Bn    8
#define Npt   4096
#define FIN   256
#define DIM   32
#define TD    256          // HEADS*DIM
#define C3    768          // 3*TD
#define NE    42
#define NPH   42
#define HW    1764         // NE*NPH
#define Pp    10
#define MLP_H 128
#define H3    16           // 2*C3/(3*DIM)
#define Mrows (Bn*HW)      // 14112 = 147*96 = 882*16
#define CATW  1536         // 2*C3
#define MT    6            // M-tiles per block in strip GEMM (96 rows)
#define KC    32           // K-chunk

typedef float v2f __attribute__((ext_vector_type(2)));
typedef float v8f __attribute__((ext_vector_type(8)));

#define WMMA_F32_16x16x4(a, b, c) \
    __builtin_amdgcn_wmma_f32_16x16x4_f32(false, (a), false, (b), (short)0, (c), false, false)

// ---------------------------------------------------------------------------
// Kernel 0: bin indices (searchsorted on eta/phi edges) + stable per-bin rank
// ---------------------------------------------------------------------------
__global__ void k_bins(const float* __restrict__ xc, const unsigned char* __restrict__ mask,
                       int* __restrict__ binrow, int* __restrict__ slot) {
    int b = blockIdx.x;
    __shared__ int gsh[Npt];
    for (int i = threadIdx.x; i < Npt; i += blockDim.x) {
        int idx = b * Npt + i;
        float eta = xc[idx * 3 + 0];
        float sp  = xc[idx * 3 + 1];
        float cp  = xc[idx * 3 + 2];
        float phi = atan2f(sp, cp);
        int ei = 0, pi = 0;
        #pragma unroll
        for (int j = 0; j < 41; j++) {
            float ee = (float)(-5.0 + 0.25 * (double)j);
            ei += (ee < eta) ? 1 : 0;
        }
        #pragma unroll
        for (int j = 0; j < 41; j++) {
            float pe = (float)(-3.14159265358979323846 + (double)j * (3.14159265358979323846 / 20.0));
            pi += (pe < phi) ? 1 : 0;
        }
        int br = ei * NPH + pi;
        binrow[idx] = br;
        gsh[i] = mask[idx] ? br : (NE * NPH);
    }
    __syncthreads();
    for (int i = threadIdx.x; i < Npt; i += blockDim.x) {
        int g = gsh[i];
        int cnt = 0;
        for (int j = 0; j < i; j++) cnt += (gsh[j] == g) ? 1 : 0;
        slot[b * Npt + i] = cnt;   // rank among same-bin points by original index
    }
}

// inv[b*HW+bin][p] = point index (or -1)
__global__ void k_inv_init(int* __restrict__ inv) {
    int i = blockIdx.x * 256 + threadIdx.x;
    if (i < Bn * HW * Pp) inv[i] = -1;
}
__global__ void k_inv_scatter(const int* __restrict__ binrow, const int* __restrict__ slot,
                              const unsigned char* __restrict__ mask, int* __restrict__ inv) {
    int idx = blockIdx.x * 256 + threadIdx.x;
    if (idx >= Bn * Npt) return;
    if (!mask[idx]) return;
    int s = slot[idx];
    if (s >= Pp) return;
    int b = idx >> 12;
    int n = idx & (Npt - 1);
    inv[(size_t)(b * HW + binrow[idx]) * Pp + s] = n;
}

// ---------------------------------------------------------------------------
// Strip GEMM: 256 threads = 8 waves; block computes 96(M) x 128(N).
// A (96 x KC) staged in LDS per K-chunk; each wave owns one 16-col N-tile and
// MT=6 M-tiles, reusing each B fragment across 6 WMMAs.
//   C[m,n] = act( sum_k A[m,k] * B[k,n] + bias[n] )
// Requires: K % KC == 0, grid.y * 96 == M, grid.x * 128 == N.
// ---------------------------------------------------------------------------
template <bool RELU, bool BIAS>
__global__ void k_gemm_strip(const float* __restrict__ A, const float* __restrict__ Bm,
                             const float* __restrict__ bias, float* __restrict__ C,
                             int K, int lda, int ldb, int ldc) {
    __shared__ float Ash[MT * 16][KC + 1];
    int tid  = threadIdx.x;
    int lane = tid & 31;
    int w    = tid >> 5;             // wave id 0..7
    int half = lane >> 4;
    int lr   = lane & 15;
    int m0   = blockIdx.y * (MT * 16);
    int n0   = blockIdx.x * 128 + w * 16;
    int n    = n0 + lr;

    int ldrow = tid >> 4;            // 0..15 (row within a pass)
    int ldk   = (tid & 15) * 2;      // even k within chunk

    v8f acc[MT];
    #pragma unroll
    for (int mt = 0; mt < MT; mt++) acc[mt] = (v8f)0.0f;

    for (int k0 = 0; k0 < K; k0 += KC) {
        // ---- cooperative A stage: 96 rows x 32 k ----
        #pragma unroll
        for (int pass = 0; pass < MT; pass++) {
            int row = pass * 16 + ldrow;
            const float* ap = A + (size_t)(m0 + row) * lda + k0 + ldk;
            Ash[row][ldk]     = ap[0];
            Ash[row][ldk + 1] = ap[1];
            __builtin_prefetch(ap + KC, 0, 0);
        }
        __syncthreads();
        // ---- 8 WMMA K-steps, B fragment reused across MT M-tiles ----
        #pragma unroll
        for (int kc = 0; kc < KC; kc += 4) {
            int kk = k0 + kc + 2 * half;
            v2f bv;
            bv.x = Bm[(size_t)kk * ldb + n];
            bv.y = Bm[(size_t)(kk + 1) * ldb + n];
            #pragma unroll
            for (int mt = 0; mt < MT; mt++) {
                v2f a;
                a.x = Ash[mt * 16 + lr][kc + 2 * half];
                a.y = Ash[mt * 16 + lr][kc + 2 * half + 1];
                acc[mt] = WMMA_F32_16x16x4(a, bv, acc[mt]);
            }
        }
        __builtin_prefetch(&Bm[(size_t)(k0 + KC) * ldb + n], 0, 0);
        __syncthreads();
    }

    float bb = BIAS ? bias[n] : 0.0f;
    #pragma unroll
    for (int mt = 0; mt < MT; mt++) {
        int mrow = m0 + mt * 16 + (half ? 8 : 0);
        #pragma unroll
        for (int r = 0; r < 8; r++) {
            float v = acc[mt][r] + bb;
            if (RELU) v = fmaxf(v, 0.0f);
            C[(size_t)(mrow + r) * ldc + n] = v;
        }
    }
}

// ---------------------------------------------------------------------------
// GEMM1 (fused gather) strip variant: flat[m,k] = x[b, inv[m][k/256], k%256] or 0
// M=14112, K=2560, N=128. Gather happens in the cooperative LDS loader.
// ---------------------------------------------------------------------------
__global__ void k_gemm1_strip(const float* __restrict__ x, const int* __restrict__ inv,
                              const float* __restrict__ W1, const float* __restrict__ b1,
                              float* __restrict__ C) {
    __shared__ float Ash[MT * 16][KC + 1];
    int tid  = threadIdx.x;
    int lane = tid & 31;
    int w    = tid >> 5;
    int half = lane >> 4;
    int lr   = lane & 15;
    int m0   = blockIdx.y * (MT * 16);
    int n    = w * 16 + lr;          // N = 128, single strip

    int ldrow = tid >> 4;
    int ldk   = (tid & 15) * 2;

    v8f acc[MT];
    #pragma unroll
    for (int mt = 0; mt < MT; mt++) acc[mt] = (v8f)0.0f;

    for (int k0 = 0; k0 < Pp * FIN; k0 += KC) {
        #pragma unroll
        for (int pass = 0; pass < MT; pass++) {
            int row = pass * 16 + ldrow;
            int m   = m0 + row;
            int k   = k0 + ldk;
            int p   = k >> 8;
            int f   = k & 255;                 // even; f+1 in same slot
            int nn  = inv[(size_t)m * Pp + p];
            float a0 = 0.0f, a1 = 0.0f;
            if (nn >= 0) {
                const float* xr = x + ((size_t)(m / HW) * Npt + nn) * FIN + f;
                a0 = xr[0];
                a1 = xr[1];
            }
            Ash[row][ldk]     = a0;
            Ash[row][ldk + 1] = a1;
        }
        __syncthreads();
        #pragma unroll
        for (int kc = 0; kc < KC; kc += 4) {
            int kk = k0 + kc + 2 * half;
            v2f bv;
            bv.x = W1[(size_t)kk * MLP_H + n];
            bv.y = W1[(size_t)(kk + 1) * MLP_H + n];
            #pragma unroll
            for (int mt = 0; mt < MT; mt++) {
                v2f a;
                a.x = Ash[mt * 16 + lr][kc + 2 * half];
                a.y = Ash[mt * 16 + lr][kc + 2 * half + 1];
                acc[mt] = WMMA_F32_16x16x4(a, bv, acc[mt]);
            }
        }
        __builtin_prefetch(&W1[(size_t)(k0 + KC) * MLP_H + n], 0, 0);
        __syncthreads();
    }

    float bb = b1[n];
    #pragma unroll
    for (int mt = 0; mt < MT; mt++) {
        int mrow = m0 + mt * 16 + (half ? 8 : 0);
        #pragma unroll
        for (int r = 0; r < 8; r++) {
            float v = fmaxf(acc[mt][r] + bb, 0.0f);   // ReLU
            C[(size_t)(mrow + r) * MLP_H + n] = v;
        }
    }
}

// ---------------------------------------------------------------------------
// Single-wave generic WMMA GEMM (kept for the 24-group pointwise, K=32,
// B transposed: B[k][n] = Bp[n*ldb + k]); z-batched via blockIdx.z.
// ---------------------------------------------------------------------------
__global__ void k_gemm_pw(const float* __restrict__ A, const float* __restrict__ Bm,
                          float* __restrict__ C,
                          int K, int lda, int ldb, int ldc,
                          int zsA, int zsB, int zsC) {
    int lane = threadIdx.x;
    int half = lane >> 4;
    int lr   = lane & 15;
    const float* Ap = A + (size_t)blockIdx.z * zsA;
    const float* Bp = Bm + (size_t)blockIdx.z * zsB;
    float*       Cp = C + (size_t)blockIdx.z * zsC;
    int m0 = blockIdx.y * 16;
    int n0 = blockIdx.x * 16;
    const float* arow = Ap + (size_t)(m0 + lr) * lda;
    const float* brow = Bp + (size_t)(n0 + lr) * ldb;   // transposed B
    v8f acc = (v8f)0.0f;
    for (int k = 2 * half; k < K; k += 4) {
        v2f a;
        a.x = arow[k];
        a.y = arow[k + 1];
        v2f bv;
        bv.x = brow[k];
        bv.y = brow[k + 1];
        acc = WMMA_F32_16x16x4(a, bv, acc);
    }
    int n = n0 + lr;
    int mrow = m0 + (half ? 8 : 0);
    #pragma unroll
    for (int r = 0; r < 8; r++) {
        Cp[(size_t)(mrow + r) * ldc + n] = acc[r];
    }
}

// ---------------------------------------------------------------------------
// Depthwise 5x5 SAME conv over (NE,NPH); input = cat[:, :768] (ld CATW)
// ---------------------------------------------------------------------------
__global__ void k_dwconv(const float* __restrict__ cat, const float* __restrict__ dww,
                         float* __restrict__ out) {
    int idx = blockIdx.x * 256 + threadIdx.x;
    if (idx >= Bn * HW * C3) return;
    int c  = idx % C3;
    int hw = (idx / C3) % HW;
    int b  = idx / (C3 * HW);
    int h = hw / NPH, w = hw % NPH;
    float acc = 0.0f;
    #pragma unroll
    for (int kh = 0; kh < 5; kh++) {
        int hh = h + kh - 2;
        if (hh < 0 || hh >= NE) continue;
        #pragma unroll
        for (int kw = 0; kw < 5; kw++) {
            int ww = w + kw - 2;
            if (ww < 0 || ww >= NPH) continue;
            acc += cat[(size_t)(b * HW + hh * NPH + ww) * CATW + c] *
                   dww[(kh * 5 + kw) * C3 + c];
        }
    }
    out[(size_t)(b * HW + hw) * C3 + c] = acc;
}

// ---------------------------------------------------------------------------
// Attention stage 1: vk[b,h,d,e] = sum_n v[d,n]*k[e,n]  (d: 0..32, row 32 = ones)
// One block per (b,h); n tiled through LDS in chunks of 12 (147*12 = 1764).
// ---------------------------------------------------------------------------
__global__ void k_vk(const float* __restrict__ cat, float* __restrict__ vkb) {
    int bh = blockIdx.x;
    int b = bh >> 4, h = bh & 15;
    int ch0 = h * 96;
    __shared__ float ksh[12 * 32];
    __shared__ float vsh[12 * 32];
    int tid = threadIdx.x;
    int d0 = tid >> 5, e0 = tid & 31;   // pairs 0..1023
    int e1 = tid;                        // pairs 1024..1055 (d=32, ones row), tid<32
    float acc0 = 0.0f, acc1 = 0.0f;
    for (int c0 = 0; c0 < HW; c0 += 12) {
        if (tid < 768) {
            int j  = tid >> 6;
            int ch = tid & 63;           // 0..31 -> k chan, 32..63 -> v chan
            int n  = c0 + j;
            float v = cat[(size_t)(b * HW + n) * CATW + ch0 + 32 + ch];
            if (ch < 32) ksh[j * 32 + ch] = fmaxf(v, 0.0f);
            else         vsh[j * 32 + (ch - 32)] = v;
        }
        __syncthreads();
        #pragma unroll
        for (int j = 0; j < 12; j++) {
            acc0 += vsh[j * 32 + d0] * ksh[j * 32 + e0];
            if (tid < 32) acc1 += ksh[j * 32 + e1];   // v==1 for padded row
        }
        __syncthreads();
    }
    vkb[(size_t)bh * 1056 + tid] = acc0;
    if (tid < 32) vkb[(size_t)bh * 1056 + 1024 + tid] = acc1;
}

// ---------------------------------------------------------------------------
// Attention stage 2: out[d,n] = (vk[d,:]@q[:,n]) / (vk[32,:]@q[:,n] + EPS)
// writes att[(b*HW+n)*512 + h*32 + d]
// ---------------------------------------------------------------------------
__global__ void k_att(const float* __restrict__ cat, const float* __restrict__ vkb,
                      float* __restrict__ att) {
    int bh = blockIdx.x;
    int b = bh >> 4, h = bh & 15;
    int ch0 = h * 96;
    __shared__ float vks[1056];
    for (int i = threadIdx.x; i < 1056; i += 256) vks[i] = vkb[(size_t)bh * 1056 + i];
    __syncthreads();
    int n = blockIdx.y * 256 + threadIdx.x;
    if (n >= HW) return;
    const float* row = cat + (size_t)(b * HW + n) * CATW + ch0;
    float q[32];
    #pragma unroll
    for (int e = 0; e < 32; e++) q[e] = fmaxf(row[e], 0.0f);
    float denom = 0.0f;
    #pragma unroll
    for (int e = 0; e < 32; e++) denom += vks[1024 + e] * q[e];
    denom += 1e-15f;
    float* orow = att + (size_t)(b * HW + n) * (H3 * DIM) + h * DIM;
    for (int d = 0; d < 32; d++) {
        float num = 0.0f;
        #pragma unroll
        for (int e = 0; e < 32; e++) num += vks[d * 32 + e] * q[e];
        orow[d] = num / denom;
    }
}

// ---------------------------------------------------------------------------
// Final: batch-norm + gather back to points (+ mask)
// ---------------------------------------------------------------------------
__global__ void k_gather(const float* __restrict__ ybuf, const int* __restrict__ binrow,
                         const unsigned char* __restrict__ mask,
                         const float* __restrict__ gamma, const float* __restrict__ beta,
                         const float* __restrict__ mean, const float* __restrict__ var,
                         float* __restrict__ out) {
    int pt = blockIdx.x;           // b*Npt + n
    int c  = threadIdx.x;          // 0..255
    int b  = pt >> 12;
    int br = binrow[pt];
    float y = ybuf[(size_t)(b * HW + br) * FIN + c];
    y = (y - mean[c]) / sqrtf(var[c] + 1e-5f) * gamma[c] + beta[c];
    out[(size_t)pt * FIN + c] = mask[pt] ? y : 0.0f;
}

// ---------------------------------------------------------------------------
extern "C" void kernel_launch(void* const* d_in, const int* in_sizes, int n_in,
                              void* d_out, int out_size, void* d_ws, size_t ws_size,
                              hipStream_t stream) {
    const float*         x    = (const float*)d_in[0];
    const float*         xc   = (const float*)d_in[1];
    const unsigned char* mask = (const unsigned char*)d_in[2];
    const float* W1    = (const float*)d_in[3];
    const float* b1    = (const float*)d_in[4];
    const float* W2    = (const float*)d_in[5];
    const float* b2    = (const float*)d_in[6];
    const float* qkvw  = (const float*)d_in[7];
    const float* dww   = (const float*)d_in[8];
    const float* pww   = (const float*)d_in[9];
    const float* projw = (const float*)d_in[10];
    const float* gam   = (const float*)d_in[11];
    const float* bet   = (const float*)d_in[12];
    const float* mu    = (const float*)d_in[13];
    const float* var   = (const float*)d_in[14];
    float* out = (float*)d_out;

    char* ws = (char*)d_ws;
    size_t off = 0;
    auto alloc = [&](size_t bytes) -> void* {
        void* p = ws + off;
        off += (bytes + 255) & ~((size_t)255);
        return p;
    };
    int*   binrow = (int*)alloc((size_t)Bn * Npt * 4);
    int*   slot   = (int*)alloc((size_t)Bn * Npt * 4);
    int*   inv    = (int*)alloc((size_t)Bn * HW * Pp * 4);
    float* hbuf   = (float*)alloc((size_t)Mrows * MLP_H * 4);
    float* agg    = (float*)alloc((size_t)Mrows * MLP_H * 4);
    float* cat    = (float*)alloc((size_t)Mrows * CATW * 4);
    float* dwb    = (float*)alloc((size_t)Mrows * C3 * 4);
    float* vkb    = (float*)alloc((size_t)Bn * H3 * 1056 * 4);
    float* att    = (float*)alloc((size_t)Mrows * H3 * DIM * 4);
    float* ybuf   = (float*)alloc((size_t)Mrows * FIN * 4);
    (void)ws_size; (void)in_sizes; (void)n_in; (void)out_size;

    const int MGRID = Mrows / (MT * 16);   // 147

    // 0) binning + stable per-bin rank
    k_bins<<<Bn, 1024, 0, stream>>>(xc, mask, binrow, slot);
    k_inv_init<<<(Bn * HW * Pp + 255) / 256, 256, 0, stream>>>(inv);
    k_inv_scatter<<<(Bn * Npt) / 256, 256, 0, stream>>>(binrow, slot, mask, inv);

    // 1) MLP layer 1 with fused scatter/gather: (14112 x 2560) @ (2560 x 128) + b1, ReLU
    k_gemm1_strip<<<dim3(1, MGRID), 256, 0, stream>>>(x, inv, W1, b1, hbuf);

    // 2) MLP layer 2: h @ W2 + b2
    k_gemm_strip<false, true><<<dim3(MLP_H / 128, MGRID), 256, 0, stream>>>(
        hbuf, W2, b2, agg, MLP_H, MLP_H, MLP_H, MLP_H);

    // 3) qkv projection: agg @ qkv_w -> cat[:, 0:768]
    k_gemm_strip<false, false><<<dim3(C3 / 128, MGRID), 256, 0, stream>>>(
        agg, qkvw, nullptr, cat, MLP_H, MLP_H, C3, CATW);

    // 4) depthwise 5x5 SAME conv on qkv -> dwb
    k_dwconv<<<(Bn * HW * C3 + 255) / 256, 256, 0, stream>>>(cat, dww, dwb);

    // 5) grouped pointwise: 24 batched GEMMs (K=32, N=32, B transposed) -> cat[:, 768:1536]
    k_gemm_pw<<<dim3(2, Mrows / 16, 24), 32, 0, stream>>>(
        dwb, pww, cat + C3, 32, C3, 32, CATW,
        /*zsA=*/32, /*zsB=*/32 * 32, /*zsC=*/32);

    // 6) linear attention
    k_vk<<<Bn * H3, 1024, 0, stream>>>(cat, vkb);
    k_att<<<dim3(Bn * H3, (HW + 255) / 256), 256, 0, stream>>>(cat, vkb, att);

    // 7) output projection: att @ proj_w -> ybuf
    k_gemm_strip<false, false><<<dim3(FIN / 128, MGRID), 256, 0, stream>>>(
        att, projw, nullptr, ybuf, H3 * DIM, H3 * DIM, FIN, FIN);

    // 8) batch-norm + gather to points + mask
    k_gather<<<Bn * Npt, FIN, 0, stream>>>(ybuf, binrow, mask, gam, bet, mu, var, out);
}